// kernel_generated_1_21887153341167
// MI455X (gfx1250) — compile-verified
//
#include <hip/hip_runtime.h>

typedef __attribute__((ext_vector_type(2))) float v2f;
typedef __attribute__((ext_vector_type(8))) float v8f;

static constexpr int HH   = 56;
static constexpr int WW   = 56;
static constexpr int HW   = HH * WW;        // 3136 (multiple of 16)
static constexpr int KTOT = 84;             // 12 channels * 7 taps
static constexpr int NCH  = 48;             // output channels per group
static constexpr int ROWS = 2 * HW;         // 6272 flattened (o,n,m) rows
static constexpr int ROW_TILES = ROWS / 16; // 392

// ---------------------------------------------------------------------------
// Prologue: fold the two einsum weights into one 48x84 matrix Wc[p][jk]
// (row-major in K so a lane's {k0,k0+1} B pair is one aligned b64 LDS load):
//   Wc[p, j*7+kk] = sum_i W1[j,kk,i] * W0[p,i]
// ---------------------------------------------------------------------------
__global__ void fold_weights_kernel(const float* __restrict__ W0,   // (48, 9)
                                    const float* __restrict__ W1,   // (12, 7, 9)
                                    float* __restrict__ Wc) {       // (48, 84)
  int idx = blockIdx.x * blockDim.x + threadIdx.x;
  if (idx >= NCH * KTOT) return;
  int p  = idx / KTOT;       // 0..47
  int jk = idx - p * KTOT;   // 0..83 == j*7 + kk (contiguous in W1)
  const float* w1 = W1 + jk * 9;
  const float* w0 = W0 + p  * 9;
  float s = 0.f;
#pragma unroll
  for (int i = 0; i < 9; ++i) s += w1[i] * w0[i];
  Wc[p * KTOT + jk] = s;
}

// ---------------------------------------------------------------------------
// Main fused kernel: im2col GEMM (M=6272, N=48, K=84) on V_WMMA_F32_16X16X4.
// Block = 128 threads = 4 waves; each wave computes one 16-row tile x all
// three 16-wide N tiles, so each A fragment feeds 3 v_wmma ops.
// ---------------------------------------------------------------------------
__global__ __launch_bounds__(128)
void conv_wmma_kernel(const float* __restrict__ x,    // viewed as (2,12,56,56)
                      const float* __restrict__ Wc,   // (48, 84)
                      float* __restrict__ out) {      // (2,48,56,56) + roll
  __shared__ alignas(16) float lWc[NCH * KTOT];       // 16128 bytes

  const int tid = threadIdx.x;
  for (int i = tid; i < NCH * KTOT; i += 128) lWc[i] = Wc[i];
  __syncthreads();

  const int wave  = tid >> 5;
  const int lane  = tid & 31;
  const int laneh = lane >> 4;   // 0: lanes 0-15, 1: lanes 16-31
  const int lanem = lane & 15;

  const int row_tile = blockIdx.x * 4 + wave;        // 0..391
  // Lane's A row (M = lanem for both lane halves, per ISA 16x4 A layout).
  const int row = row_tile * 16 + lanem;
  const int o   = row / HW;                          // constant per tile
  const int rem = row - o * HW;
  const int n   = rem / WW;
  const int m   = rem - n * WW;
  const float* xb = x + (o * 12) * HW + n * WW;      // + j*HW + mm
  __builtin_prefetch(xb + m, 0, 3);                  // global_prefetch_b8

  // A VGPR0 holds K = khalf, VGPR1 holds K = khalf+1 (khalf = 0 or 2);
  // B mirrors this: lanes 0-15 carry rows 4b+0/4b+1, lanes 16-31 rows +2/+3.
  const int khalf = laneh * 2;

  // Incremental (j, kk) decomposition of k = 4*b + khalf (+1); stride 4 < 7
  // so each step wraps at most once: no division in the loop.
  int j0 = 0, kk0 = khalf;
  int j1 = 0, kk1 = khalf + 1;

  // B pointer for this lane's N column: pair {Wc[p][k0], Wc[p][k0+1]}
  // is contiguous & 8B-aligned (k0 even, row stride 84 even) -> ds_load_b64.
  const float* w0r = &lWc[( 0 + lanem) * KTOT + khalf];
  const float* w1r = &lWc[(16 + lanem) * KTOT + khalf];
  const float* w2r = &lWc[(32 + lanem) * KTOT + khalf];

  v8f acc0 = {};
  v8f acc1 = {};
  v8f acc2 = {};

  for (int b = 0; b < KTOT / 4; ++b) {   // 21 K-steps of 4
    // im2col gather with width padding (predicated -> v_cndmask, EXEC all-1s)
    const int mm0 = m + kk0 - 3;
    const int mm1 = m + kk1 - 3;
    const float a0 = (mm0 >= 0 && mm0 < WW) ? xb[j0 * HW + mm0] : 0.f;
    const float a1 = (mm1 >= 0 && mm1 < WW) ? xb[j1 * HW + mm1] : 0.f;
    v2f a = {a0, a1};

    const v2f b0 = *(const v2f*)(w0r + b * 4);
    const v2f b1 = *(const v2f*)(w1r + b * 4);
    const v2f b2 = *(const v2f*)(w2r + b * 4);

    acc0 = __builtin_amdgcn_wmma_f32_16x16x4_f32(false, a, false, b0,
                                                 (short)0, acc0, false, false);
    acc1 = __builtin_amdgcn_wmma_f32_16x16x4_f32(false, a, false, b1,
                                                 (short)0, acc1, false, false);
    acc2 = __builtin_amdgcn_wmma_f32_16x16x4_f32(false, a, false, b2,
                                                 (short)0, acc2, false, false);

    // k += 4 with single-wrap carry into j
    kk0 += 4; if (kk0 >= 7) { kk0 -= 7; ++j0; }
    kk1 += 4; if (kk1 >= 7) { kk1 -= 7; ++j1; }
  }

  // Store 16x16 tiles; C/D layout: VGPR r -> M = r + 8*laneh, N = lanem.
  // Fold in roll(+1 on H): t5[o,p,n,m] -> out[o,p,(n+1)%56,m].
  // o is tile-constant; (n2, m2) advance by increment-with-carry.
  const int grow0 = row_tile * 16 + laneh * 8;       // row of acc[...][0]
  const int rem0  = grow0 - o * HW;
  int n2 = rem0 / WW;
  int m2 = rem0 - n2 * WW;
  int nr = n2 + 1; if (nr == HH) nr = 0;
  float* obase = out + (size_t)(o * NCH) * HW;

#pragma unroll
  for (int r = 0; r < 8; ++r) {
    float* ob = obase + nr * WW + m2;
    ob[(size_t)( 0 + lanem) * HW] = acc0[r];
    ob[(size_t)(16 + lanem) * HW] = acc1[r];
    ob[(size_t)(32 + lanem) * HW] = acc2[r];
    // advance one flattened row: m2++, carry into the rolled row index
    ++m2;
    if (m2 == WW) { m2 = 0; ++nr; if (nr == HH) nr = 0; }
  }
}

extern "C" void kernel_launch(void* const* d_in, const int* in_sizes, int n_in,
                              void* d_out, int out_size, void* d_ws, size_t ws_size,
                              hipStream_t stream) {
  const float* x  = (const float*)d_in[0];   // (1,24,56,56) == (2,12,56,56)
  const float* W0 = (const float*)d_in[1];   // (48,9)
  const float* W1 = (const float*)d_in[2];   // (12,7,9)
  float* out = (float*)d_out;                // (1,96,56,56)
  float* Wc  = (float*)d_ws;                 // 48*84 floats = 16128 B

  fold_weights_kernel<<<(NCH * KTOT + 127) / 128, 128, 0, stream>>>(W0, W1, Wc);
  conv_wmma_kernel<<<ROW_TILES / 4, 128, 0, stream>>>(x, Wc, out);
}